// HashEncoder_24000277250548
// MI455X (gfx1250) — compile-verified
//
#include <hip/hip_runtime.h>
#include <math.h>

typedef float v2f __attribute__((ext_vector_type(2)));
typedef float v4f __attribute__((ext_vector_type(4)));

#define NLVL 16
// With the reference constants (BASE_RES=16, per_level_scale=1.3195079565048218,
// MAX_PARAMS=2^19) levels 0..5 are dense (res^3 <= padded size, res<=65) and
// levels 6..15 are hashed (res>=85 -> res^3 > 2^19). This split is a property of
// the fixed constants, so it is resolved at compile time.
#define N_DENSE 6
#define PRIME_Y 2654435761u
#define PRIME_Z 805459861u

struct Meta {
  const v2f*   base[NLVL];   // table + 2*OFFSETS[l], as float2*
  float        scale[NLVL];  // float32(BASE_RES*exp(l*ln(S)) - 1)
  unsigned int rm[NLVL];     // dense: resolution; hashed: map_size-1 (2^19-1)
};

// Warm the L2 with the full parameter table (~46 MB < 192 MB L2).
// Emits gfx1250 global_prefetch_b8 (no VGPR return, no counter).
__global__ __launch_bounds__(256) void hashenc_warm_kernel(const float* __restrict__ t,
                                                           unsigned n_lines) {
  unsigned i = blockIdx.x * blockDim.x + threadIdx.x;
  if (i < n_lines) {
    __builtin_prefetch(t + (size_t)i * 32u, 0, 1);  // one 128B line per lane
  }
}

template <bool DENSE>
__device__ __forceinline__ void enc_level(const v2f* __restrict__ base, float s,
                                          unsigned rm, float px, float py, float pz,
                                          float& o0, float& o1) {
  // Mirror reference: f32 mul then add (no contraction -> identical floor boundaries).
  const float gx = __fadd_rn(__fmul_rn(px, s), 0.5f);
  const float gy = __fadd_rn(__fmul_rn(py, s), 0.5f);
  const float gz = __fadd_rn(__fmul_rn(pz, s), 0.5f);
  const float fx = floorf(gx), fy = floorf(gy), fz = floorf(gz);
  const float rx = gx - fx, ry = gy - fy, rz = gz - fz;
  const unsigned x0 = (unsigned)fx, y0 = (unsigned)fy, z0 = (unsigned)fz;
  const unsigned x1 = x0 + 1u;

  unsigned idx[8];
  if (DENSE) {
    const unsigned R    = rm;
    const unsigned R2   = R * R;
    const unsigned b000 = x0 + y0 * R + z0 * R2;
    idx[0] = b000;          idx[1] = b000 + 1u;
    idx[2] = b000 + R;      idx[3] = b000 + R + 1u;
    idx[4] = b000 + R2;     idx[5] = b000 + R2 + 1u;
    idx[6] = b000 + R2 + R; idx[7] = b000 + R2 + R + 1u;
  } else {
    const unsigned m   = rm;
    const unsigned hy0 = y0 * PRIME_Y, hy1 = hy0 + PRIME_Y;
    const unsigned hz0 = z0 * PRIME_Z, hz1 = hz0 + PRIME_Z;
    const unsigned e00 = hy0 ^ hz0, e10 = hy1 ^ hz0;
    const unsigned e01 = hy0 ^ hz1, e11 = hy1 ^ hz1;
    idx[0] = (x0 ^ e00) & m;  idx[1] = (x1 ^ e00) & m;
    idx[2] = (x0 ^ e10) & m;  idx[3] = (x1 ^ e10) & m;
    idx[4] = (x0 ^ e01) & m;  idx[5] = (x1 ^ e01) & m;
    idx[6] = (x0 ^ e11) & m;  idx[7] = (x1 ^ e11) & m;
  }

  // All 8 gathers issued before use: one global_load_b64 clause, max MLP vs L2 latency.
  v2f f[8];
#pragma unroll
  for (int c = 0; c < 8; ++c) f[c] = base[idx[c]];

  const float wx[2] = {1.0f - rx, rx};
  const float wy[2] = {1.0f - ry, ry};
  const float wz[2] = {1.0f - rz, rz};

  // Packed accumulate: both features of a corner sit in one float2, so each corner
  // is a single v_pk_fma_f32 (VOP3P) with the scalar weight splat via OPSEL.
  v2f acc = {0.0f, 0.0f};
#pragma unroll
  for (int c = 0; c < 8; ++c) {
    // ((wx*wy)*wz): same association as the reference; the 4 wx*wy pairs CSE.
    const float w = (wx[c & 1] * wy[(c >> 1) & 1]) * wz[(c >> 2) & 1];
    const v2f w2 = {w, w};
    acc = __builtin_elementwise_fma(w2, f[c], acc);  // per-component fused, same order
  }
  o0 = acc.x;
  o1 = acc.y;
}

__global__ __launch_bounds__(256) void hashenc_kernel(const float* __restrict__ positions,
                                                      float* __restrict__ out,
                                                      Meta meta, int n_points) {
  int b = blockIdx.x * blockDim.x + threadIdx.x;
  if (b >= n_points) return;

  // Streaming, read-once: NT loads keep L2 reserved for the table.
  const float px = __builtin_nontemporal_load(positions + 3 * (size_t)b + 0);
  const float py = __builtin_nontemporal_load(positions + 3 * (size_t)b + 1);
  const float pz = __builtin_nontemporal_load(positions + 3 * (size_t)b + 2);

  float out32[32];

#pragma unroll
  for (int l = 0; l < N_DENSE; ++l)
    enc_level<true>(meta.base[l], meta.scale[l], meta.rm[l], px, py, pz,
                    out32[2 * l], out32[2 * l + 1]);
#pragma unroll
  for (int l = N_DENSE; l < NLVL; ++l)
    enc_level<false>(meta.base[l], meta.scale[l], meta.rm[l], px, py, pz,
                     out32[2 * l], out32[2 * l + 1]);

  // 128 contiguous bytes per thread -> each wave32 writes 4KB contiguous.
  // Non-temporal: the 256 MB output stream must not evict the table from L2.
  v4f* __restrict__ o4 = (v4f*)(out + 32 * (size_t)b);
#pragma unroll
  for (int i = 0; i < 8; ++i) {
    v4f v = {out32[4 * i + 0], out32[4 * i + 1], out32[4 * i + 2], out32[4 * i + 3]};
    __builtin_nontemporal_store(v, o4 + i);
  }
}

extern "C" void kernel_launch(void* const* d_in, const int* in_sizes, int n_in,
                              void* d_out, int out_size, void* d_ws, size_t ws_size,
                              hipStream_t stream) {
  const float* positions = (const float*)d_in[0];
  const float* table     = (const float*)d_in[1];
  float*       out       = (float*)d_out;

  const int n_points = in_sizes[0] / 3;

  // Replicate reference _level_meta()/_encode() metadata in double precision.
  Meta meta;
  long long off = 0;  // in table entries (float2 units)
  const double lnS = log(1.3195079565048218);
  for (int i = 0; i < NLVL; ++i) {
    const double sd = 16.0 * exp((double)i * lnS) - 1.0;  // scale (double)
    const float  sf = (float)sd;                          // np.float32(scale)
    const long long res_meta = (long long)ceil(sd) + 1;   // _level_meta res (double path)
    const long long cube = res_meta * res_meta * res_meta;
    long long p = cube;
    if (p % 8) p = ((p + 7) / 8) * 8;
    if (p > (1LL << 19)) p = (1LL << 19);
    const bool dense = (cube <= p);

    meta.base[i]  = (const v2f*)table + off;              // folds OFFSETS[l] into saddr
    meta.scale[i] = sf;
    meta.rm[i]    = dense ? ((unsigned)ceilf(sf) + 1u)    // _encode resolution (f32 path)
                          : (unsigned)(p - 1);            // power-of-two hash mask
    off += p;
  }

  // 1) Pull the whole table into L2 (one global_prefetch_b8 per 128B line).
  const unsigned n_lines = (unsigned)(((size_t)in_sizes[1] * sizeof(float) + 127) / 128);
  hashenc_warm_kernel<<<(n_lines + 255) / 256, 256, 0, stream>>>(table, n_lines);

  // 2) Main gather/interp kernel: one thread per point, 128 outstanding b64 gathers.
  const int tpb = 256;
  const int blocks = (n_points + tpb - 1) / tpb;
  hashenc_kernel<<<blocks, tpb, 0, stream>>>(positions, out, meta, n_points);
}